// KineticMatrixLayer_17360257811181
// MI455X (gfx1250) — compile-verified
//
#include <hip/hip_runtime.h>
#include <hip/hip_bf16.h>
#include <stdint.h>

// ---------------------------------------------------------------------------
// Top-3 eigenvalues of 12x12 reversible-chain kinetic matrices.
//
// K is tridiagonal with positive off-diagonals -> similar to a symmetric
// tridiagonal S with the same diagonal d_i = -(f_i + b_{i-1}) and
// off-diagonal e_i = sqrt(f_i * b_i). Eigenvalues are real; we find the
// 3 largest (ascending indices 11,10,9) via Sturm-count bisection using
// e_i^2 = f_i*b_i directly (no sqrt needed).
//
// Data movement (MI455X): 5.8 MB in + 0.8 MB out -> ~0.28 us at 23.3 TB/s.
// Per-lane rates are a contiguous 88B row => stage each block's contiguous
// 22528 B slab into LDS with coalesced CDNA5 async global->LDS b64 loads
// (11 per thread, uniform fast path, no per-lane predication), fence with
// s_wait_asynccnt + barrier, then read per-lane (dword stride 22 across
// lanes is bank-conflict-free mod 64).
// ---------------------------------------------------------------------------

#define NUM_RATES   22
#define NUM_STATES  12
#define TOP_K       3
#define TPB         256                    // 8 wave32s; 1 element per lane
#define LDS_FLOATS  (TPB * NUM_RATES)      // 5632 dwords = 22528 B of LDS
#define B64_PER_THR (LDS_FLOATS / (2 * TPB))   // 11 x b64 per thread, exact

__global__ __launch_bounds__(TPB)
void kinetic_topk_eig_kernel(const float* __restrict__ rates,
                             float* __restrict__ out,
                             int batch)
{
    __shared__ float lds[LDS_FLOATS];

    const int tid       = threadIdx.x;
    const int base_elem = blockIdx.x * TPB;

    // ---- Stage block slab global -> LDS via CDNA5 async-to-LDS loads ----
    {
        const int nelem = min(TPB, batch - base_elem);        // scalar (uniform)
        const uint32_t lds_base = (uint32_t)(uintptr_t)(&lds[0]); // low 32 bits = LDS offset
        const uint64_t gbase = (uint64_t)(uintptr_t)(rates + (size_t)base_elem * NUM_RATES);

        if (nelem == TPB) {
            // Full block: 11 x b64 per thread, back-to-back, no predication.
            #pragma unroll
            for (int j = 0; j < B64_PER_THR; ++j) {
                const uint32_t boff = (uint32_t)(j * TPB + tid) * 8u; // byte offset, coalesced
                const uint32_t loff = lds_base + boff;
                asm volatile("global_load_async_to_lds_b64 %0, %1, %2"
                             :
                             : "v"(loff), "v"(boff), "s"(gbase)
                             : "memory");
            }
        } else {
            // Tail block: guarded b64 pairs (ndw is always even: NUM_RATES=22).
            const uint32_t ndw = (uint32_t)(nelem * NUM_RATES);
            #pragma unroll
            for (int j = 0; j < B64_PER_THR; ++j) {
                const uint32_t dw = 2u * (uint32_t)(j * TPB + tid);
                if (dw < ndw) {
                    const uint32_t boff = dw * 4u;
                    const uint32_t loff = lds_base + boff;
                    asm volatile("global_load_async_to_lds_b64 %0, %1, %2"
                                 :
                                 : "v"(loff), "v"(boff), "s"(gbase)
                                 : "memory");
                }
            }
        }
        asm volatile("s_wait_asynccnt 0" ::: "memory");
    }
    __syncthreads();

    // ---- Build symmetric tridiagonal (d, e^2) from this lane's 22 rates ----
    float f[NUM_STATES - 1], b[NUM_STATES - 1];
    {
        const float* r = &lds[tid * NUM_RATES];
        #pragma unroll
        for (int i = 0; i < NUM_STATES - 1; ++i) {
            f[i] = r[i];                    // forward rate i  (i -> i+1)
            b[i] = r[NUM_STATES - 1 + i];   // backward rate i (i+1 -> i)
        }
    }

    float dgn[NUM_STATES], e2[NUM_STATES - 1];
    #pragma unroll
    for (int i = 0; i < NUM_STATES; ++i) {
        float d = 0.0f;
        if (i < NUM_STATES - 1) d -= f[i];
        if (i > 0)              d -= b[i - 1];
        dgn[i] = d;
    }
    #pragma unroll
    for (int i = 0; i < NUM_STATES - 1; ++i) e2[i] = f[i] * b[i];

    // ---- Bisection for eigenvalues #11, #10, #9 (ascending index) ----
    // Spectrum is contained in [-4, 0] (Gershgorin; rates in [0,1)).
    const float PIVMIN = 1e-20f;
    float res[TOP_K];

    #pragma unroll
    for (int k = 0; k < TOP_K; ++k) {
        const int need = NUM_STATES - k;    // count(x) >= need  <=>  x > lambda_{11-k}
        float lo = -4.25f;
        float hi = 3.0e-4f;
        #pragma unroll 1
        for (int it = 0; it < 30; ++it) {
            const float x = 0.5f * (lo + hi);
            // Sturm count: number of eigenvalues < x
            int cnt;
            {
                float q = dgn[0] - x;
                cnt = (q < 0.0f) ? 1 : 0;
                #pragma unroll
                for (int i = 1; i < NUM_STATES; ++i) {
                    float qs = (fabsf(q) < PIVMIN) ? ((q < 0.0f) ? -PIVMIN : PIVMIN) : q;
                    q = (dgn[i] - x) - e2[i - 1] * __builtin_amdgcn_rcpf(qs);
                    cnt += (q < 0.0f) ? 1 : 0;
                }
            }
            const bool above = (cnt >= need);   // x is above the target eigenvalue
            hi = above ? x : hi;
            lo = above ? lo : x;
        }
        res[k] = 0.5f * (lo + hi);
    }

    // ---- Write top-3 (descending, matching lax.top_k) ----
    const int e = base_elem + tid;
    if (e < batch) {
        out[e * TOP_K + 0] = res[0];
        out[e * TOP_K + 1] = res[1];
        out[e * TOP_K + 2] = res[2];
    }
}

extern "C" void kernel_launch(void* const* d_in, const int* in_sizes, int n_in,
                              void* d_out, int out_size, void* d_ws, size_t ws_size,
                              hipStream_t stream)
{
    const float* rates = (const float*)d_in[0];
    float* out = (float*)d_out;
    const int batch = in_sizes[0] / NUM_RATES;
    const int grid  = (batch + TPB - 1) / TPB;
    kinetic_topk_eig_kernel<<<grid, TPB, 0, stream>>>(rates, out, batch);
}